// PointcloudGrouping_14078902796515
// MI455X (gfx1250) — compile-verified
//
#include <hip/hip_runtime.h>
#include <hip/hip_bf16.h>

#define NUM_GROUPS 512
#define GROUP_SIZE 32
#define GROUP_RADIUS 0.1f
#define UPSCALE 512
#define CONTEXT 256
#define BB 8
#define NN 8192

typedef __attribute__((ext_vector_type(2))) float v2f;
typedef __attribute__((ext_vector_type(8))) float v8f;
typedef __attribute__((ext_vector_type(4))) int   v4i;

// ---------------------------------------------------------------------------
// Kernel 1: Farthest Point Sampling. One 1024-thread workgroup per batch.
// mind[] lives entirely in LDS; 512 sequential steps of (distance update +
// block-wide argmax). Centers written with |c|^2 in .w so the WMMA ball-query
// kernel gets its per-row threshold for free.
// NOTE: start index is a deterministic hash stand-in for jax threefry randint.
// ---------------------------------------------------------------------------
__global__ __launch_bounds__(1024) void fps_kernel(const float4* __restrict__ pts,
                                                   const int* __restrict__ lengths,
                                                   float4* __restrict__ centersWS)
{
    __shared__ float mind[NN];
    __shared__ float redV[1024];
    __shared__ int   redI[1024];
    __shared__ int   s_last;
    const int b = blockIdx.x, tid = threadIdx.x;
    const int length = lengths[b];

    unsigned h = 1103515245u * (unsigned)(b + 1) + 12345u;
    h ^= h >> 13; h *= 2654435761u; h ^= h >> 16;
    const int start = (int)(h % (unsigned)length);

    for (int j = 0; j < NN / 1024; ++j) {
        int n = tid + j * 1024;
        mind[n] = (n < length) ? 1e10f : -1.0f;
    }
    if (tid == 0) s_last = start;
    __syncthreads();

    const float4* pb = pts + (size_t)b * NN;
    for (int k = 0; k < NUM_GROUPS; ++k) {
        const int last = s_last;
        const float4 lp = pb[last];
        if (tid == 0)
            centersWS[b * NUM_GROUPS + k] =
                make_float4(lp.x, lp.y, lp.z, lp.x*lp.x + lp.y*lp.y + lp.z*lp.z);

        float bestV = -2.0f; int bestI = NN;
        for (int j = 0; j < NN / 1024; ++j) {
            int n = tid + j * 1024;
            float4 p = pb[n];
            float dx = p.x - lp.x, dy = p.y - lp.y, dz = p.z - lp.z;
            float d = dx*dx + dy*dy + dz*dz;
            float m = mind[n];
            float nm = (m < 0.0f) ? -1.0f : fminf(m, d);
            mind[n] = nm;
            if (nm > bestV || (nm == bestV && n < bestI)) { bestV = nm; bestI = n; }
        }
        redV[tid] = bestV; redI[tid] = bestI;
        __syncthreads();
        for (int s = 512; s > 0; s >>= 1) {
            if (tid < s) {
                float v = redV[tid + s]; int i2 = redI[tid + s];
                if (v > redV[tid] || (v == redV[tid] && i2 < redI[tid])) {
                    redV[tid] = v; redI[tid] = i2;
                }
            }
            __syncthreads();
        }
        if (tid == 0) s_last = redI[0];
        __syncthreads();
    }
}

// ---------------------------------------------------------------------------
// Kernel 2: Ball query via V_WMMA_F32_16X16X4_F32.
// acc = A(16x4) x B(4x16): A row m = [cx,cy,cz,1], B col n = [-2px,-2py,-2pz,|p|^2]
// => acc = -2 c.p + |p|^2 ; within iff acc < r^2 - |c|^2.
// Each of the 8 C-rows is reduced to a 32-bit wave ballot whose bit layout is
// already {row r+8 [31:16], row r [15:0]}, so we store the raw ballot word:
//   mask32[((b*32 + gtile)*8 + r)*512 + ntile]
// All 8 stores live in ONE lane==0 block (single exec-mask toggle, clause-able).
// ---------------------------------------------------------------------------
__global__ __launch_bounds__(256) void ballq_wmma_kernel(const float4* __restrict__ pts,
                                                         const int* __restrict__ lengths,
                                                         const float4* __restrict__ centersWS,
                                                         unsigned* __restrict__ mask32)
{
    __shared__ float4 sCC[16];
    const int b = blockIdx.y;
    const int gtile = blockIdx.x;
    const int g0 = gtile * 16;
    const int tid = threadIdx.x;
    const int lane = tid & 31, wave = tid >> 5;
    if (tid < 16) sCC[tid] = centersWS[b * NUM_GROUPS + g0 + tid];
    __syncthreads();

    const int length = lengths[b];
    const int hi = lane >> 4;            // 0: lanes 0-15, 1: lanes 16-31
    const float4 ca = sCC[lane & 15];
    const float aA[4] = { ca.x, ca.y, ca.z, 1.0f };
    v2f A; A.x = aA[hi * 2]; A.y = aA[hi * 2 + 1];

    const float R2 = GROUP_RADIUS * GROUP_RADIUS;
    float thr[8];
#pragma unroll
    for (int r = 0; r < 8; ++r) thr[r] = R2 - sCC[r + (hi << 3)].w;  // r^2 - |c|^2

    const float4* pb = pts + (size_t)b * NN;
    unsigned* mb = mask32 + (size_t)((b * 32 + gtile) * 8) * (NN / 16);

    for (int nt = wave; nt < NN / 16; nt += 8) {
        const int col = nt * 16 + (lane & 15);
        __builtin_prefetch(&pb[col + 256], 0, 1);   // global_prefetch_b8
        float4 p = pb[col];
        float pp = p.x*p.x + p.y*p.y + p.z*p.z;
        if (col >= length) pp += 1e10f;
        v2f Bv;
        Bv.x = hi ? (-2.0f * p.z) : (-2.0f * p.x);
        Bv.y = hi ? pp            : (-2.0f * p.y);
        v8f acc = {};
        acc = __builtin_amdgcn_wmma_f32_16x16x4_f32(false, A, false, Bv,
                                                    (short)0, acc, false, false);
        unsigned mm[8];
#pragma unroll
        for (int r = 0; r < 8; ++r)
            mm[r] = __builtin_amdgcn_ballot_w32(acc[r] < thr[r]);
        if (lane == 0) {
#pragma unroll
            for (int r = 0; r < 8; ++r)
                mb[(size_t)r * (NN / 16) + nt] = mm[r];
        }
    }
}

// ---------------------------------------------------------------------------
// Kernel 3: per-(batch,group) candidate compaction (index-ordered, capped at
// UPSCALE=512) + 512-wide LDS bitonic sort by energy (descending, stable),
// emit groups / point_mask for g < CONTEXT and the "group full" flag.
// The 2KB mask row is staged into LDS with the gfx1250 async copy engine
// (GLOBAL_LOAD_ASYNC_TO_LDS_B128 + s_wait_asynccnt). The builtin's signature
// (learned from the round-2 diagnostic) takes a global int4* then an LDS ptr.
// ---------------------------------------------------------------------------
__global__ __launch_bounds__(256) void group_select_kernel(const float4* __restrict__ pts,
                                                           const float4* __restrict__ centersWS,
                                                           const unsigned* __restrict__ mask32,
                                                           int* __restrict__ fullWS,
                                                           float* __restrict__ out_groups,
                                                           float* __restrict__ out_pmask)
{
    __shared__ __attribute__((aligned(16))) unsigned s_mask[NN / 16];
    __shared__ int   s_cand[UPSCALE];
    __shared__ float s_e[UPSCALE];
    __shared__ int   s_ix[UPSCALE];
    __shared__ int   s_scan[256];

    const int bg = blockIdx.x;
    const int b = bg >> 9, g = bg & 511;
    const int tid = threadIdx.x;

    const int gtile = g >> 4, r = g & 7, hi16 = (g >> 3) & 1;
    const unsigned* mb = mask32 + (size_t)((b * 32 + gtile) * 8 + r) * (NN / 16);

#if defined(__has_builtin) && __has_builtin(__builtin_amdgcn_global_load_async_to_lds_b128)
    if (tid < (NN / 16) / 4) {   // 128 lanes x 16B = 2KB
        typedef __attribute__((address_space(1))) v4i* gv4i_p;
        typedef __attribute__((address_space(3))) v4i* lv4i_p;
        __builtin_amdgcn_global_load_async_to_lds_b128(
            (gv4i_p)(mb + tid * 4), (lv4i_p)(&s_mask[tid * 4]), 0, 0);
    }
#if defined(__has_builtin) && __has_builtin(__builtin_amdgcn_s_wait_asynccnt)
    __builtin_amdgcn_s_wait_asynccnt(0);
#endif
#else
    for (int i = tid; i < NN / 16; i += 256) s_mask[i] = mb[i];
#endif
    __syncthreads();

    int base = 0;
    for (int chunk = 0; chunk < 2; ++chunk) {
        const int t = chunk * 256 + tid;
        unsigned m = (s_mask[t] >> (hi16 * 16)) & 0xFFFFu;
        const int pc = __popc(m);
        s_scan[tid] = pc;
        __syncthreads();
        for (int off = 1; off < 256; off <<= 1) {
            int v = (tid >= off) ? s_scan[tid - off] : 0;
            __syncthreads();
            s_scan[tid] += v;
            __syncthreads();
        }
        const int excl = s_scan[tid] - pc;
        const int total = s_scan[255];
        int p = base + excl;
        while (m) {                       // ascending bit -> ascending point index
            int bit = __ffs(m) - 1;
            if (p < UPSCALE) s_cand[p] = t * 16 + bit;
            p++;
            m &= m - 1;
        }
        base += total;
        __syncthreads();
    }
    const int cnt = min(base, UPSCALE);

    const float4* pb = pts + (size_t)b * NN;
    for (int i = tid; i < UPSCALE; i += 256) {
        s_e[i] = (i < cnt) ? pb[s_cand[i]].w : -INFINITY;
        s_ix[i] = i;
    }
    __syncthreads();

    // bitonic sort: energy descending, tie -> smaller original candidate pos
    for (int k = 2; k <= UPSCALE; k <<= 1) {
        for (int j = k >> 1; j > 0; j >>= 1) {
            for (int t = tid; t < UPSCALE; t += 256) {
                const int l = t ^ j;
                if (l > t) {
                    const bool asc = ((t & k) == 0);
                    float ea = s_e[t]; int ia = s_ix[t];
                    float eb = s_e[l]; int ib = s_ix[l];
                    bool b_before_a = (eb > ea) || (eb == ea && ib < ia);
                    bool a_before_b = (ea > eb) || (ea == eb && ia < ib);
                    if (asc ? b_before_a : a_before_b) {
                        s_e[t] = eb; s_ix[t] = ib;
                        s_e[l] = ea; s_ix[l] = ia;
                    }
                }
            }
            __syncthreads();
        }
    }

    if (tid == 0) fullWS[bg] = (cnt >= GROUP_SIZE) ? 1 : 0;

    if (g < CONTEXT && tid < GROUP_SIZE) {
        const int k = tid;
        int selpos;
        if (k < cnt)      selpos = s_ix[k];
        else if (cnt > 0) selpos = s_ix[0];   // idx_f = idx[:, :, :1] fallback
        else              selpos = -1;        // no candidates -> point 0
        const int n = (selpos >= 0) ? s_cand[selpos] : 0;
        const float4 p = pb[n];
        const float4 c = centersWS[b * NUM_GROUPS + g];
        const size_t o = ((size_t)(b * CONTEXT + g) * GROUP_SIZE + k) * 4;
        const float inv = 1.0f / GROUP_RADIUS;
        out_groups[o + 0] = (p.x - c.x) * inv;
        out_groups[o + 1] = (p.y - c.y) * inv;
        out_groups[o + 2] = (p.z - c.z) * inv;
        out_groups[o + 3] = p.w;
        const int pl = min(cnt, GROUP_SIZE);
        out_pmask[(size_t)(b * CONTEXT + g) * GROUP_SIZE + k] = (k < pl) ? 1.0f : 0.0f;
    }
}

// ---------------------------------------------------------------------------
// Kernel 4: group_lengths = sum of full flags over all 512 groups -> prefix
// embedding_mask; also emit centers[:, :CONTEXT, :3].
// ---------------------------------------------------------------------------
__global__ __launch_bounds__(512) void finalize_kernel(const float4* __restrict__ centersWS,
                                                       const int* __restrict__ fullWS,
                                                       float* __restrict__ out_centers,
                                                       float* __restrict__ out_emb)
{
    __shared__ int s[512];
    const int b = blockIdx.x, g = threadIdx.x;
    s[g] = fullWS[b * NUM_GROUPS + g];
    __syncthreads();
    for (int st = 256; st > 0; st >>= 1) {
        if (g < st) s[g] += s[g + st];
        __syncthreads();
    }
    const int gl = s[0];
    if (g < CONTEXT) {
        out_emb[b * CONTEXT + g] = (g < gl) ? 1.0f : 0.0f;
        const float4 c = centersWS[b * NUM_GROUPS + g];
        const size_t o = (size_t)(b * CONTEXT + g) * 3;
        out_centers[o + 0] = c.x;
        out_centers[o + 1] = c.y;
        out_centers[o + 2] = c.z;
    }
}

extern "C" void kernel_launch(void* const* d_in, const int* in_sizes, int n_in,
                              void* d_out, int out_size, void* d_ws, size_t ws_size,
                              hipStream_t stream) {
    const float4* pts     = (const float4*)d_in[0];   // (8, 8192, 4) f32
    const int*    lengths = (const int*)d_in[1];      // (8,) i32

    char* ws = (char*)d_ws;
    float4*   centersWS = (float4*)ws;                      //  64 KB: (8,512) xyz+|c|^2
    int*      fullWS    = (int*)(ws + 64 * 1024);           //  16 KB: (8,512) full flags
    unsigned* mask32    = (unsigned*)(ws + 80 * 1024);      //   4 MB: (8,32,8,512) ballot words

    float* out_groups  = (float*)d_out;                     // (8,256,32,4)
    float* out_centers = out_groups + BB * CONTEXT * GROUP_SIZE * 4;
    float* out_emb     = out_centers + BB * CONTEXT * 3;
    float* out_pmask   = out_emb + BB * CONTEXT;

    fps_kernel<<<BB, 1024, 0, stream>>>(pts, lengths, centersWS);
    ballq_wmma_kernel<<<dim3(NUM_GROUPS / 16, BB), 256, 0, stream>>>(pts, lengths, centersWS, mask32);
    group_select_kernel<<<BB * NUM_GROUPS, 256, 0, stream>>>(pts, centersWS, mask32, fullWS,
                                                             out_groups, out_pmask);
    finalize_kernel<<<BB, 512, 0, stream>>>(centersWS, fullWS, out_centers, out_emb);
}